// ShallowSGCNNet_30803505446917
// MI455X (gfx1250) — compile-verified
//
#include <hip/hip_runtime.h>
#include <math.h>

typedef __attribute__((ext_vector_type(2))) float v2f;
typedef __attribute__((ext_vector_type(8))) float v8f;
typedef __attribute__((ext_vector_type(4))) int v4i;

// ---- workspace layout (float offsets) ----
#define WS_STAT1_SUM 0         // 20 (padded 32)
#define WS_STAT1_SQ  32
#define WS_SCALE1    64
#define WS_SHIFT1    96
#define WS_A2        128       // 32x32
#define WS_STAT2_SUM 1152
#define WS_STAT2_SQ  1184
#define WS_W2        1280      // 4x640 folded fc weights
#define WS_BIAS2     3840      // 4
#define WS_M         4096      // (B=256, NK=20, C=32) = 163840
#define WS_XP        167936    // pooled conv out (256,20,32,188)

static __device__ __forceinline__ v8f wmma4(v2f a, v2f b, v8f c) {
  // V_WMMA_F32_16X16X4_F32 : D(16x16,f32) = A(16x4,f32) x B(4x16,f32) + C
  return __builtin_amdgcn_wmma_f32_16x16x4_f32(false, a, false, b, (short)0, c,
                                               false, false);
}
static __device__ __forceinline__ float elu1(float x) {
  return x > 0.f ? x : expm1f(x);
}

// ---- async global->LDS staging (GLOBAL_LOAD_ASYNC_TO_LDS_B128, ASYNCcnt) ----
#if __has_builtin(__builtin_amdgcn_global_load_async_to_lds_b128)
#define HAS_ASYNC_LDS 1
typedef __attribute__((address_space(1))) v4i* g4p;  // global int4*
typedef __attribute__((address_space(3))) v4i* l4p;  // LDS int4*
static __device__ __forceinline__ void async_copy16(const void* g, void* l) {
  __builtin_amdgcn_global_load_async_to_lds_b128((g4p)g, (l4p)l, 0, 0);
}
static __device__ __forceinline__ void async_wait0() {
#if __has_builtin(__builtin_amdgcn_s_wait_asynccnt)
  __builtin_amdgcn_s_wait_asynccnt(0);
#else
  asm volatile("s_wait_asynccnt 0x0" ::: "memory");
#endif
}
#else
#define HAS_ASYNC_LDS 0
#endif

// ---------------- K0: zero the stats / small scratch region ----------------
__global__ void k0_zero(float* __restrict__ ws) {
  int i = blockIdx.x * 256 + threadIdx.x;
  if (i < 4096) ws[i] = 0.f;
}

// ---------------- K1: conv(1x25) + bias + ELU + bn1 stats + pair-pool ------
// One block per (b,c) row. GEMM view: out[t,k] = sum_s in[t+s] * w[k,s]
// M-tile = 16 consecutive t, K padded 25->28 (7 wmma k-steps), N padded 20->32
__global__ __launch_bounds__(128) void k1_conv(const float* __restrict__ in,
                                               const float* __restrict__ cw,
                                               const float* __restrict__ cb,
                                               float* __restrict__ ws) {
  __shared__ float sIn[416];        // input row, zero padded
  __shared__ float sB[28][32];      // weights: [s][k], zero padded
  __shared__ float sSum[32], sSq[32];
  const int b = blockIdx.x >> 5;
  const int c = blockIdx.x & 31;
  const int tid = threadIdx.x;

  for (int i = tid; i < 416; i += 128)
    sIn[i] = (i < 400) ? in[(size_t)(b * 32 + c) * 400 + i] : 0.f;
  for (int i = tid; i < 28 * 32; i += 128) {
    int s = i >> 5, k = i & 31;
    sB[s][k] = (s < 25 && k < 20) ? cw[k * 25 + s] : 0.f;
  }
  if (tid < 32) { sSum[tid] = 0.f; sSq[tid] = 0.f; }
  __syncthreads();

  const int wave = tid >> 5, lane = tid & 31;
  const int half = (lane < 16) ? 0 : 1;     // which half of the wave
  const int n = lane & 15;                  // N (= output channel) within tile
  const int k0 = n, k1 = n + 16;
  const bool k1ok = (n < 4);                // channels 16..19 valid
  float ls0 = 0.f, lq0 = 0.f, ls1 = 0.f, lq1 = 0.f;
  const float bias0 = cb[k0];
  const float bias1 = k1ok ? cb[k1] : 0.f;
  float* xp = ws + WS_XP;

  for (int i = 0; i < 6; ++i) {             // 24 M-tiles / 4 waves
    const int mt = wave + 4 * i;
    const int t0 = mt * 16;
    v8f a0 = {}, a1 = {};
#pragma unroll
    for (int kc = 0; kc < 7; ++kc) {
      const int kb = 4 * kc + 2 * half;     // A layout: lanes 16-31 hold K=2,3
      v2f af, b0, b1;
      af.x = sIn[t0 + n + kb];  af.y = sIn[t0 + n + kb + 1];
      b0.x = sB[kb][n];         b0.y = sB[kb + 1][n];
      b1.x = sB[kb][n + 16];    b1.y = sB[kb + 1][n + 16];
      a0 = wmma4(af, b0, a0);
      a1 = wmma4(af, b1, a1);
    }
    float e0[8], e1[8];
#pragma unroll
    for (int r = 0; r < 8; ++r) {           // D: vgpr r = row r (+8 for hi half)
      const int t = t0 + 8 * half + r;
      const bool tv = (t < 376);
      float v0 = elu1(a0[r] + bias0);
      float v1 = elu1(a1[r] + bias1);
      e0[r] = v0; e1[r] = v1;
      if (tv) { ls0 += v0; lq0 += v0 * v0;
                if (k1ok) { ls1 += v1; lq1 += v1 * v1; } }
    }
    // BN is affine per-k => commutes with pair pooling: pool raw ELU output.
    const int t2b = t0 / 2 + 4 * half;
    if (t2b < 188) {
      float4 p0;
      p0.x = 0.5f * (e0[0] + e0[1]); p0.y = 0.5f * (e0[2] + e0[3]);
      p0.z = 0.5f * (e0[4] + e0[5]); p0.w = 0.5f * (e0[6] + e0[7]);
      *(float4*)(xp + ((size_t)(b * 20 + k0) * 32 + c) * 188 + t2b) = p0;
      if (k1ok) {
        float4 p1;
        p1.x = 0.5f * (e1[0] + e1[1]); p1.y = 0.5f * (e1[2] + e1[3]);
        p1.z = 0.5f * (e1[4] + e1[5]); p1.w = 0.5f * (e1[6] + e1[7]);
        *(float4*)(xp + ((size_t)(b * 20 + k1) * 32 + c) * 188 + t2b) = p1;
      }
    }
  }
  atomicAdd(&sSum[k0], ls0); atomicAdd(&sSq[k0], lq0);
  if (k1ok) { atomicAdd(&sSum[k1], ls1); atomicAdd(&sSq[k1], lq1); }
  __syncthreads();
  if (tid < 20) {
    atomicAdd(&ws[WS_STAT1_SUM + tid], sSum[tid]);
    atomicAdd(&ws[WS_STAT1_SQ + tid], sSq[tid]);
  }
}

// ---------------- K2: bn1 affine + adjacency A and A^2 ---------------------
__global__ __launch_bounds__(64) void k2_prep(const float* __restrict__ ew,
                                              const float* __restrict__ g1,
                                              const float* __restrict__ b1,
                                              float* __restrict__ ws) {
  __shared__ float dis[32];
  __shared__ float sA[32][33];
  const int tid = threadIdx.x;
  if (tid < 32) {
    float d = 0.f;
    for (int r = 0; r < 32; ++r) {
      float w = ew[r * 32 + tid];           // col[e]=e%32 => deg[c]=sum_r w[r*32+c]
      if (w >= 0.1f) d += w;
    }
    dis[tid] = (d > 0.f) ? rsqrtf(d) : 0.f;
  }
  if (tid < 20) {
    const float N1 = 256.f * 32.f * 376.f;
    float mean = ws[WS_STAT1_SUM + tid] / N1;
    float var = ws[WS_STAT1_SQ + tid] / N1 - mean * mean;
    float sc = g1[tid] * rsqrtf(var + 1e-5f);
    ws[WS_SCALE1 + tid] = sc;
    ws[WS_SHIFT1 + tid] = b1[tid] - mean * sc;
  }
  __syncthreads();
  for (int nidx = tid; nidx < 1024; nidx += 64) {
    const int i = nidx >> 5, j = nidx & 31;  // A[i=col][j=row]
    float w = ew[j * 32 + i];
    w = (w >= 0.1f) ? w : 0.f;
    sA[i][j] = dis[j] * w * dis[i];
  }
  __syncthreads();
  for (int nidx = tid; nidx < 1024; nidx += 64) {  // A2 = A @ A (two hops in one)
    const int i = nidx >> 5, j = nidx & 31;
    float s = 0.f;
    for (int p = 0; p < 32; ++p) s += sA[i][p] * sA[p][j];
    ws[WS_A2 + nidx] = s;
  }
}

// ---------------- K3: per-(b,k): Y = A2 @ X ; Z = Y @ linW^T ; ELU; stats; mean
// X tiles (24KB each) are double-buffered into LDS with async global->LDS
// copies so pair pi+1's HBM traffic overlaps pair pi's WMMA work.
// bn1 affine is folded into the Y store: Y = sc*(A2@Xraw) + sh*rowsum(A2).
__global__ __launch_bounds__(256) void k3_graph(const float* __restrict__ linw,
                                                const float* __restrict__ linb,
                                                float* __restrict__ ws) {
  __shared__ float sA2[32][36];
  __shared__ float sXd[2][32][196]; // raw X double buffer; cols 188..195 zero
  __shared__ float sY[32][196];
  __shared__ float sLW[188][32];    // B for step2: [f][o], o padded 20->32
  __shared__ float sRS[32];         // row sums of A2
  __shared__ float sStat[2];
  const int tid = threadIdx.x;
  const int wave = tid >> 5, lane = tid & 31;
  const int khi = (lane < 16) ? 0 : 2;      // A/B frag K base shift for hi half
  const int pbase = blockIdx.x * 4;

#if HAS_ASYNC_LDS
  {  // kick off pair 0's tile before doing anything else
    const float* xp0 = ws + WS_XP + (size_t)pbase * 32 * 188;
    for (int q = tid; q < 32 * 47; q += 256) {
      const int cc = q / 47, fq = q - cc * 47;
      async_copy16(xp0 + cc * 188 + fq * 4, &sXd[0][cc][fq * 4]);
    }
  }
#endif
  // zero the pad columns of both X buffers (async copies never touch them)
  for (int idx = tid; idx < 512; idx += 256) {
    const int bf = idx >> 8, r = (idx >> 3) & 31, f = 188 + (idx & 7);
    sXd[bf][r][f] = 0.f;
  }
  for (int idx = tid; idx < 1024; idx += 256)
    sA2[idx >> 5][idx & 31] = ws[WS_A2 + idx];
  for (int idx = tid; idx < 188 * 32; idx += 256) {
    const int f = idx >> 5, o = idx & 31;
    sLW[f][o] = (o < 20) ? linw[o * 188 + f] : 0.f;
  }
  __syncthreads();
  if (tid < 32) {
    float s = 0.f;
    for (int j = 0; j < 32; ++j) s += sA2[tid][j];
    sRS[tid] = s;
  }

  for (int pi = 0; pi < 4; ++pi) {
    const int p = pbase + pi;                // p = b*20 + k
    const int k = p % 20;
    const int cur = pi & 1;
    const float sc = ws[WS_SCALE1 + k], sh = ws[WS_SHIFT1 + k];
#if HAS_ASYNC_LDS
    async_wait0();                           // buf[cur] landed (per-wave)
    __syncthreads();
    if (tid == 0) { sStat[0] = 0.f; sStat[1] = 0.f; }
    if (pi < 3) {                            // overlap next tile with compute
      const float* xpn = ws + WS_XP + (size_t)(p + 1) * 32 * 188;
      for (int q = tid; q < 32 * 47; q += 256) {
        const int cc = q / 47, fq = q - cc * 47;
        async_copy16(xpn + cc * 188 + fq * 4, &sXd[cur ^ 1][cc][fq * 4]);
      }
    }
#else
    __syncthreads();
    if (tid == 0) { sStat[0] = 0.f; sStat[1] = 0.f; }
    {
      const float* xp = ws + WS_XP + (size_t)p * 32 * 188;
      for (int idx = tid; idx < 32 * 188; idx += 256) {
        const int cc = idx / 188, f = idx - cc * 188;
        sXd[cur][cc][f] = xp[idx];
      }
    }
    __syncthreads();
#endif

    // step 1: Y(32x188) = A2(32x32) @ Xraw; bn1 affine folded at store
    for (int tt = wave; tt < 24; tt += 8) {
      const int mt = tt / 12, nt = tt % 12;
      const int m = mt * 16 + (lane & 15);
      const int nn = nt * 16 + (lane & 15);
      v8f acc = {};
#pragma unroll
      for (int kc = 0; kc < 8; ++kc) {
        const int kb = 4 * kc + khi;
        v2f af, bf;
        af.x = sA2[m][kb];       af.y = sA2[m][kb + 1];
        bf.x = sXd[cur][kb][nn]; bf.y = sXd[cur][kb + 1][nn];
        acc = wmma4(af, bf, acc);
      }
      const int rb = mt * 16 + ((lane < 16) ? 0 : 8);
#pragma unroll
      for (int r = 0; r < 8; ++r)
        sY[rb + r][nn] = sc * acc[r] + sh * sRS[rb + r];
    }
    __syncthreads();

    // step 2: Z(32x20) = Y(32x188) @ linW^T(188x20); 2x2 tiles on waves 0..3
    if (wave < 4) {
      const int mt = wave >> 1, nt = wave & 1;
      const int m = mt * 16 + (lane & 15);
      const int o = nt * 16 + (lane & 15);
      v8f acc = {};
      for (int kc = 0; kc < 47; ++kc) {      // 188 = 47*4 exactly
        const int kb = 4 * kc + khi;
        v2f af, bf;
        af.x = sY[m][kb];  af.y = sY[m][kb + 1];
        bf.x = sLW[kb][o]; bf.y = sLW[kb + 1][o];
        acc = wmma4(af, bf, acc);
      }
      const bool ov = (o < 20);
      const float lb = ov ? linb[o] : 0.f;
      const int rb = mt * 16 + ((lane < 16) ? 0 : 8);
      float lsum = 0.f, lsq = 0.f;
#pragma unroll
      for (int r = 0; r < 8; ++r) {
        float v = elu1(acc[r] + lb);
        if (ov) { lsum += v; lsq += v * v; }
        sXd[cur][rb + r][o] = ov ? v : 0.f;  // stage Z (reuse dead X buffer)
      }
      atomicAdd(&sStat[0], lsum);
      atomicAdd(&sStat[1], lsq);
    }
    __syncthreads();
    if (tid < 32) {                          // m[b,k,c] = mean over o of Z
      float s = 0.f;
      for (int o = 0; o < 20; ++o) s += sXd[cur][tid][o];
      ws[WS_M + (size_t)p * 32 + tid] = s * (1.f / 20.f);
    }
    if (tid == 0) {
      atomicAdd(&ws[WS_STAT2_SUM + k], sStat[0]);
      atomicAdd(&ws[WS_STAT2_SQ + k], sStat[1]);
    }
  }
}

// ---------------- K4: fold bn2 affine + mean-over-o into fc ----------------
__global__ __launch_bounds__(64) void k4_fold(const float* __restrict__ g2,
                                              const float* __restrict__ b2,
                                              const float* __restrict__ fcw,
                                              const float* __restrict__ fcb,
                                              float* __restrict__ ws) {
  __shared__ float sa[20], sc[20];
  const int tid = threadIdx.x;
  if (tid < 20) {
    const float N2 = 256.f * 32.f * 20.f;
    float mean = ws[WS_STAT2_SUM + tid] / N2;
    float var = ws[WS_STAT2_SQ + tid] / N2 - mean * mean;
    float a = g2[tid] * rsqrtf(var + 1e-5f);
    sa[tid] = a;
    sc[tid] = b2[tid] - mean * a;
  }
  __syncthreads();
  for (int idx = tid; idx < 2560; idx += 64) {
    const int kc = idx % 640;
    ws[WS_W2 + idx] = fcw[idx] * sa[kc >> 5];
  }
  if (tid < 4) {
    float s = fcb[tid];
    for (int kc = 0; kc < 640; ++kc) s += fcw[tid * 640 + kc] * sc[kc >> 5];
    ws[WS_BIAS2 + tid] = s;
  }
}

// ---------------- K5: out[b,o] = w2[o,:] . m[b,:] + bias2[o] ---------------
__global__ __launch_bounds__(128) void k5_fc(const float* __restrict__ ws,
                                             float* __restrict__ out) {
  __shared__ float sred[4][128];
  const int b = blockIdx.x, tid = threadIdx.x;
  float a0 = 0.f, a1 = 0.f, a2 = 0.f, a3 = 0.f;
  const float* m = ws + WS_M + (size_t)b * 640;
  const float* w2 = ws + WS_W2;
  for (int kc = tid; kc < 640; kc += 128) {
    float mm = m[kc];
    a0 += w2[kc] * mm;        a1 += w2[640 + kc] * mm;
    a2 += w2[1280 + kc] * mm; a3 += w2[1920 + kc] * mm;
  }
  sred[0][tid] = a0; sred[1][tid] = a1; sred[2][tid] = a2; sred[3][tid] = a3;
  __syncthreads();
  for (int st = 64; st > 0; st >>= 1) {
    if (tid < st)
      for (int o = 0; o < 4; ++o) sred[o][tid] += sred[o][tid + st];
    __syncthreads();
  }
  if (tid < 4) out[b * 4 + tid] = sred[tid][0] + ws[WS_BIAS2 + tid];
}

extern "C" void kernel_launch(void* const* d_in, const int* in_sizes, int n_in,
                              void* d_out, int out_size, void* d_ws,
                              size_t ws_size, hipStream_t stream) {
  (void)in_sizes; (void)n_in; (void)out_size; (void)ws_size;
  const float* input = (const float*)d_in[0];
  // d_in[1] = edge_index (int32): full CxC grid (row=e/32, col=e%32) — implied.
  const float* conv_w = (const float*)d_in[2];
  const float* conv_b = (const float*)d_in[3];
  const float* bn1_g  = (const float*)d_in[4];
  const float* bn1_b  = (const float*)d_in[5];
  const float* edge_w = (const float*)d_in[6];
  const float* lin_w  = (const float*)d_in[7];
  const float* lin_b  = (const float*)d_in[8];
  const float* bn2_g  = (const float*)d_in[9];
  const float* bn2_b  = (const float*)d_in[10];
  const float* fc_w   = (const float*)d_in[11];
  const float* fc_b   = (const float*)d_in[12];
  float* ws = (float*)d_ws;
  float* out = (float*)d_out;

  k0_zero<<<16, 256, 0, stream>>>(ws);
  k1_conv<<<256 * 32, 128, 0, stream>>>(input, conv_w, conv_b, ws);
  k2_prep<<<1, 64, 0, stream>>>(edge_w, bn1_g, bn1_b, ws);
  k3_graph<<<1280, 256, 0, stream>>>(lin_w, lin_b, ws);
  k4_fold<<<1, 64, 0, stream>>>(bn2_g, bn2_b, fc_w, fc_b, ws);
  k5_fc<<<256, 128, 0, stream>>>(ws, out);
}